// MMDiscrepancyLayer_29377576305361
// MI455X (gfx1250) — compile-verified
//
#include <hip/hip_runtime.h>
#include <hip/hip_bf16.h>

typedef float v2f __attribute__((ext_vector_type(2)));
typedef float v8f __attribute__((ext_vector_type(8)));

#define N_PTS 8192
#define DIM   64
#define INV_D2 (1.0f / 4096.0f)   // 1/D^2, D=64
#define COL_CHUNKS 4
#define NT_PER_BLK (N_PTS / 16 / COL_CHUNKS)   // 128 subtiles of 16 cols

// ---------------------------------------------------------------------------
// Kernel 1: per-row squared norms for z (which=0) and true_samples (which=1)
// ---------------------------------------------------------------------------
__global__ __launch_bounds__(256) void row_norms_kernel(
    const float* __restrict__ z, const float* __restrict__ t,
    float* __restrict__ norms) {
  int gid   = blockIdx.x * blockDim.x + threadIdx.x;  // 0..16383
  int row   = gid & (N_PTS - 1);
  int which = gid >> 13;
  const float* src = which ? t : z;
  const float4* p4 = (const float4*)(src + (size_t)row * DIM);
  float s = 0.f;
#pragma unroll
  for (int i = 0; i < DIM / 4; ++i) {
    float4 v = p4[i];
    s += v.x * v.x + v.y * v.y + v.z * v.z + v.w * v.w;
  }
  norms[which * N_PTS + row] = s;
}

// ---------------------------------------------------------------------------
// Helpers for the ping-pong pipelined Gram-tile loop
// ---------------------------------------------------------------------------
__device__ __forceinline__ void load_bfrag(v2f (&b)[16], const float* brow) {
#pragma unroll
  for (int tk = 0; tk < 16; ++tk)
    b[tk] = *(const v2f*)(brow + 4 * tk);
}

__device__ __forceinline__ void tile_compute(const v2f (&a)[16], const v2f (&b)[16],
                                             const float (&xs)[8], float yv,
                                             float& acc) {
  v8f c = {};  // zero accumulator
#pragma unroll
  for (int tk = 0; tk < 16; ++tk)
    c = __builtin_amdgcn_wmma_f32_16x16x4_f32(
        /*neg_a=*/false, a[tk], /*neg_b=*/false, b[tk],
        /*c_mod=*/(short)0, c, /*reuse_a=*/false, /*reuse_b=*/false);
  // C layout: VGPR r, lane l -> M = r + 8*(l>>4), N = l&15.
#pragma unroll
  for (int r = 0; r < 8; ++r) {
    float sq = xs[r] + yv - 2.0f * c[r];
    acc += __expf(-sq * INV_D2);
  }
}

// ---------------------------------------------------------------------------
// Kernel 2: fused Gram-tile (f32 WMMA) + exp + partial sum.
// Ping-pong double-buffered: load b1(nt+1) while computing b0(nt), then swap.
// grid = (64 row-groups, 4 col-chunks, 3 pairs); 256 threads = 8 wave32.
// blockIdx.z = pair: 0 -> (z,z), 1 -> (t,t), 2 -> (z,t)
// ---------------------------------------------------------------------------
__global__ __launch_bounds__(256) void mmd_tile_kernel(
    const float* __restrict__ z, const float* __restrict__ t,
    const float* __restrict__ norms, float* __restrict__ partials) {
  const int pair = blockIdx.z;
  const float* X  = (pair == 1) ? t : z;
  const float* Y  = (pair == 0) ? z : t;
  const float* Xn = norms + ((pair == 1) ? N_PTS : 0);
  const float* Yn = norms + ((pair == 0) ? 0 : N_PTS);

  const int wave = threadIdx.x >> 5;  // 0..7
  const int lane = threadIdx.x & 31;
  const int lo   = lane & 15;
  const int hi   = lane >> 4;         // 0 or 1

  const int row_base = blockIdx.x * 128 + wave * 16;

  // Preload A fragments (reused across all column tiles).
  // ISA 32-bit A 16x4 layout: lane lo-half holds K={4t,4t+1}, hi-half K={4t+2,4t+3}.
  v2f a[16];
  const float* arow = X + (size_t)(row_base + lo) * DIM + 2 * hi;
#pragma unroll
  for (int tk = 0; tk < 16; ++tk)
    a[tk] = *(const v2f*)(arow + 4 * tk);

  // x-norms in C-fragment order: VGPR r, lane-half hi -> M = r + 8*hi.
  float xs[8];
#pragma unroll
  for (int r = 0; r < 8; ++r)
    xs[r] = Xn[row_base + r + 8 * hi];

  float acc = 0.f;

  const int nt0 = blockIdx.y * NT_PER_BLK;
  const float* ybase = Y + (size_t)lo * DIM + 2 * hi;  // + col_base*DIM per tile

  // ---- ping-pong pipeline: two physical buffers, no register copies ----
  v2f b0[16], b1[16];
  float yv0, yv1;

  load_bfrag(b0, ybase + (size_t)(nt0 * 16) * DIM);
  yv0 = Yn[nt0 * 16 + lo];

  int nt = nt0;
  for (int j = 0; j < NT_PER_BLK / 2 - 1; ++j, nt += 2) {
    load_bfrag(b1, ybase + (size_t)((nt + 1) * 16) * DIM);
    yv1 = Yn[(nt + 1) * 16 + lo];
    __builtin_amdgcn_sched_barrier(0);      // keep b1 loads above compute(b0)
    tile_compute(a, b0, xs, yv0, acc);

    load_bfrag(b0, ybase + (size_t)((nt + 2) * 16) * DIM);
    yv0 = Yn[(nt + 2) * 16 + lo];
    __builtin_amdgcn_sched_barrier(0);      // keep b0 loads above compute(b1)
    tile_compute(a, b1, xs, yv1, acc);
  }
  // tail: last two tiles (nt, nt+1)
  load_bfrag(b1, ybase + (size_t)((nt + 1) * 16) * DIM);
  yv1 = Yn[(nt + 1) * 16 + lo];
  __builtin_amdgcn_sched_barrier(0);
  tile_compute(a, b0, xs, yv0, acc);
  tile_compute(a, b1, xs, yv1, acc);

  // Deterministic reduction: wave shuffle -> LDS -> one partial per block.
#pragma unroll
  for (int off = 16; off > 0; off >>= 1)
    acc += __shfl_xor(acc, off, 32);

  __shared__ float sm[8];
  if (lane == 0) sm[wave] = acc;
  __syncthreads();
  if (threadIdx.x == 0) {
    float s = 0.f;
#pragma unroll
    for (int w = 0; w < 8; ++w) s += sm[w];
    partials[pair * (64 * COL_CHUNKS) + blockIdx.y * 64 + blockIdx.x] = s;
  }
}

// ---------------------------------------------------------------------------
// Kernel 3: combine 3*256 partials: (S_zz + S_tt - 2*S_zt) / N^2
// ---------------------------------------------------------------------------
__global__ __launch_bounds__(256) void mmd_reduce_kernel(
    const float* __restrict__ partials, float* __restrict__ out) {
  const int i    = threadIdx.x;  // 0..255
  const int wave = i >> 5;
  const int lane = i & 31;

  float v = partials[i] + partials[256 + i] - 2.0f * partials[512 + i];
#pragma unroll
  for (int off = 16; off > 0; off >>= 1)
    v += __shfl_xor(v, off, 32);

  __shared__ float sm[8];
  if (lane == 0) sm[wave] = v;
  __syncthreads();
  if (i == 0) {
    float s = 0.f;
#pragma unroll
    for (int w = 0; w < 8; ++w) s += sm[w];
    out[0] = s / ((float)N_PTS * (float)N_PTS);
  }
}

// ---------------------------------------------------------------------------
extern "C" void kernel_launch(void* const* d_in, const int* in_sizes, int n_in,
                              void* d_out, int out_size, void* d_ws, size_t ws_size,
                              hipStream_t stream) {
  const float* z = (const float*)d_in[0];
  const float* t = (const float*)d_in[1];

  float* norms    = (float*)d_ws;       // 2 * 8192 floats
  float* partials = norms + 2 * N_PTS;  // 3 * 256 floats
  float* out      = (float*)d_out;

  row_norms_kernel<<<64, 256, 0, stream>>>(z, t, norms);

  dim3 grid(64, COL_CHUNKS, 3);
  mmd_tile_kernel<<<grid, 256, 0, stream>>>(z, t, norms, partials);

  mmd_reduce_kernel<<<1, 256, 0, stream>>>(partials, out);
}